// Double_GCN_29463475650636
// MI455X (gfx1250) — compile-verified
//
#include <hip/hip_runtime.h>
#include <hip/hip_bf16.h>
#include <math.h>

typedef __attribute__((ext_vector_type(2))) float v2f;
typedef __attribute__((ext_vector_type(8))) float v8f;

#define D_G   16384
#define BATCH 128
#define NNZ_T 262144
#define CL1F  32
#define CL1K  6
#define POOL1 8
#define VP1   (D_G / POOL1)          // 2048
#define FC1FIN (CL1F * VP1)          // 65536
#define SUMFIN 896
#define OUTC  10

// ---------------------------------------------------------------------------
// 1) transpose x_in [B, V] -> T0 [V, B]
// ---------------------------------------------------------------------------
__global__ void k_transpose_x0(const float* __restrict__ x_in, float* __restrict__ t0) {
    int g = blockIdx.x * 256 + threadIdx.x;          // g = v*128 + b
    if (g >= D_G * BATCH) return;
    int v = g >> 7;
    int b = g & 127;
    t0[g] = x_in[(size_t)b * D_G + v];
}

// ---------------------------------------------------------------------------
// 2) SpMM: y[r,:] += v * x[c,:]   (rows sorted; register accumulate, flush on
//    row change with global_atomic_add_f32). lane = batch column (128 thr/blk)
// ---------------------------------------------------------------------------
#define SPMM_CHUNK 128
__global__ void k_spmm(const int* __restrict__ rows, const int* __restrict__ cols,
                       const float* __restrict__ vals,
                       const float* __restrict__ x, float* __restrict__ y) {
    const int b = threadIdx.x;                       // 0..127
    const int base = blockIdx.x * SPMM_CHUNK;
    float acc = 0.f;
    int prev = -1;
    for (int i = 0; i < SPMM_CHUNK; ++i) {
        int idx = base + i;
        int r = rows[idx];                           // uniform -> scalar loads
        int c = cols[idx];
        float v = vals[idx];
        if (r != prev) {
            if (prev >= 0) atomicAdd(&y[(size_t)prev * BATCH + b], acc);
            acc = 0.f;
            prev = r;
        }
        acc += v * x[(size_t)c * BATCH + b];         // coalesced 512B gather
    }
    if (prev >= 0) atomicAdd(&y[(size_t)prev * BATCH + b], acc);
}

// y = 2*y - x (Chebyshev recursion combine)
__global__ void k_cheby_combine(float* __restrict__ y, const float* __restrict__ x, int n) {
    int g = blockIdx.x * 256 + threadIdx.x;
    if (g < n) y[g] = 2.f * y[g] - x[g];
}

// ---------------------------------------------------------------------------
// 3) fused cl1 GEMM (K=6) + ReLU + maxpool(8) -> xp[b, vp*32+f]
// ---------------------------------------------------------------------------
__global__ void k_cl1_pool(const float* __restrict__ t0, const float* __restrict__ t1,
                           const float* __restrict__ t2, const float* __restrict__ t3,
                           const float* __restrict__ t4, const float* __restrict__ t5,
                           const float* __restrict__ W,  const float* __restrict__ bias,
                           float* __restrict__ xp) {
    const float* T[CL1K] = {t0, t1, t2, t3, t4, t5};
    int g = blockIdx.x * 256 + threadIdx.x;          // 2^23 threads
    int f  = g & 31;
    int vp = (g >> 5) & (VP1 - 1);
    int b  = g >> 16;
    float wl[CL1K];
#pragma unroll
    for (int k = 0; k < CL1K; ++k) wl[k] = W[f * CL1K + k];
    float bv = bias[f];
    float m = 0.f;                                   // max(relu(.)) == max(0, max(pre))
#pragma unroll
    for (int vi = 0; vi < POOL1; ++vi) {
        size_t v = (size_t)vp * POOL1 + vi;
        float pre = bv;
#pragma unroll
        for (int k = 0; k < CL1K; ++k)
            pre += T[k][v * BATCH + b] * wl[k];      // wave-broadcast load
        m = fmaxf(m, pre);
    }
    xp[(size_t)b * FC1FIN + vp * CL1F + f] = m;      // coalesced store
}

// ---------------------------------------------------------------------------
// 4) fp32 WMMA GEMM: C[128,N] = act(A[128,K] @ W[N,K]^T + bias)
//    block = 256 thr (8 waves). Tile: M=128, N=64, Kc=32, double-buffered LDS,
//    register-staged global loads (one barrier/chunk). Each wave: 16(M)x64(N)
//    via 4 accumulators (A-frag reused across 4 WMMAs). Optional split-K with
//    float-atomic reduction (bias/act applied by k_bias_act afterwards).
// ---------------------------------------------------------------------------
__global__ void k_gemm128(const float* __restrict__ A, const float* __restrict__ W,
                          const float* __restrict__ bias, float* __restrict__ C,
                          int K, int N, int do_relu, int kPerBlock, int splitk) {
    __shared__ float lA[2][128][33];                 // 33.8 KB
    __shared__ float lB[2][64][33];                  // 16.9 KB
    const int tid  = threadIdx.x;
    const int lane = tid & 31;
    const int wave = tid >> 5;                       // 0..7 -> M tile
    const int n0   = blockIdx.x * 64;
    const int nl   = lane & 15;
    const int kSel = (lane >> 4) * 2;

    const int kStart = blockIdx.y * kPerBlock;
    const int nChunks = kPerBlock / 32;

    // A staging: 4 x float4 per thread
    const int segA = tid & 7;                        // 8 segments of 4 floats
    const int rbA  = tid >> 3;                       // 0..31 (+r*32)
    // B staging: 2 x float4 per thread (row n, 8 floats)
    const int nB   = tid >> 2;                       // 0..63
    const int ksB  = (tid & 3) * 8;
    const int gnB  = n0 + nB;
    const float nmask = (gnB < N) ? 1.f : 0.f;
    const float* Wrow = W + (size_t)((gnB < N) ? gnB : (N - 1)) * K;

    v8f acc[4] = {};
    float4 va[4], vb0, vb1;

    // preload chunk 0
    {
        int k0 = kStart;
#pragma unroll
        for (int r = 0; r < 4; ++r)
            va[r] = *(const float4*)(A + (size_t)(rbA + r * 32) * K + k0 + segA * 4);
        const float* wp = Wrow + k0 + ksB;
        vb0 = *(const float4*)(wp);
        vb1 = *(const float4*)(wp + 4);
    }

    int buf = 0;
    for (int ci = 0; ci < nChunks; ++ci) {
        // commit staged registers to LDS (scalar stores: padded stride)
#pragma unroll
        for (int r = 0; r < 4; ++r) {
            int row = rbA + r * 32;
            lA[buf][row][segA * 4 + 0] = va[r].x;
            lA[buf][row][segA * 4 + 1] = va[r].y;
            lA[buf][row][segA * 4 + 2] = va[r].z;
            lA[buf][row][segA * 4 + 3] = va[r].w;
        }
        lB[buf][nB][ksB + 0] = vb0.x * nmask;
        lB[buf][nB][ksB + 1] = vb0.y * nmask;
        lB[buf][nB][ksB + 2] = vb0.z * nmask;
        lB[buf][nB][ksB + 3] = vb0.w * nmask;
        lB[buf][nB][ksB + 4] = vb1.x * nmask;
        lB[buf][nB][ksB + 5] = vb1.y * nmask;
        lB[buf][nB][ksB + 6] = vb1.z * nmask;
        lB[buf][nB][ksB + 7] = vb1.w * nmask;
        __syncthreads();                             // single barrier per chunk

        // prefetch next chunk while computing this one
        if (ci + 1 < nChunks) {
            int k0 = kStart + (ci + 1) * 32;
#pragma unroll
            for (int r = 0; r < 4; ++r)
                va[r] = *(const float4*)(A + (size_t)(rbA + r * 32) * K + k0 + segA * 4);
            const float* wp = Wrow + k0 + ksB;
            vb0 = *(const float4*)(wp);
            vb1 = *(const float4*)(wp + 4);
        }

#pragma unroll
        for (int kk = 0; kk < 32; kk += 4) {
            int kb = kk + kSel;
            v2f af;
            af.x = lA[buf][wave * 16 + nl][kb];
            af.y = lA[buf][wave * 16 + nl][kb + 1];
#pragma unroll
            for (int j = 0; j < 4; ++j) {
                v2f bf;
                bf.x = lB[buf][j * 16 + nl][kb];
                bf.y = lB[buf][j * 16 + nl][kb + 1];
                acc[j] = __builtin_amdgcn_wmma_f32_16x16x4_f32(
                    false, af, false, bf, (short)0, acc[j], false, false);
            }
        }
        buf ^= 1;                                    // double-buffer: safe w/ 1 barrier
    }

#pragma unroll
    for (int j = 0; j < 4; ++j) {
        int gn = n0 + j * 16 + nl;
        if (gn < N) {
            if (!splitk) {
                float bv = bias[gn];
#pragma unroll
                for (int i = 0; i < 8; ++i) {
                    int m = wave * 16 + ((lane < 16) ? i : i + 8);
                    float v = acc[j][i] + bv;
                    if (do_relu) v = fmaxf(v, 0.f);
                    C[(size_t)m * N + gn] = v;
                }
            } else {
#pragma unroll
                for (int i = 0; i < 8; ++i) {
                    int m = wave * 16 + ((lane < 16) ? i : i + 8);
                    atomicAdd(&C[(size_t)m * N + gn], acc[j][i]);
                }
            }
        }
    }
}

// bias + optional relu finalize for split-K outputs
__global__ void k_bias_act(float* __restrict__ C, const float* __restrict__ bias,
                           int total, int N, int do_relu) {
    int g = blockIdx.x * 256 + threadIdx.x;
    if (g >= total) return;
    float v = C[g] + bias[g % N];
    if (do_relu) v = fmaxf(v, 0.f);
    C[g] = v;
}

// ---------------------------------------------------------------------------
// 5) whole GCN4 + NN4 path (batch-invariant) in one block
// ---------------------------------------------------------------------------
__global__ void k_gcn4(const float* __restrict__ x_embed, const float* __restrict__ L_mg,
                       const float* __restrict__ cl4_W,  const float* __restrict__ cl4_b,
                       const float* __restrict__ fcg1_W, const float* __restrict__ fcg1_b,
                       const float* __restrict__ fcg2_W, const float* __restrict__ fcg2_b,
                       const float* __restrict__ nn14_W, const float* __restrict__ nn14_b,
                       const float* __restrict__ nn24_W, const float* __restrict__ nn24_b,
                       float* __restrict__ g4v, float* __restrict__ n4v) {
    __shared__ float emb[640];       // [v*64+f] == x_embed layout
    __shared__ float Te[4][640];
    __shared__ float Lm[100];
    __shared__ float xg[320];        // [v*32+fo]
    __shared__ float g4[96];
    __shared__ float h1[128];
    __shared__ float hn1[256];
    const int t = threadIdx.x;

    if (t < 640) { emb[t] = x_embed[t]; Te[0][t] = emb[t]; }
    if (t < 100) Lm[t] = L_mg[t];
    __syncthreads();
    if (t < 640) {                                   // T1 = L @ T0
        int v = t >> 6, f = t & 63;
        float s = 0.f;
#pragma unroll
        for (int u = 0; u < 10; ++u) s += Lm[v * 10 + u] * Te[0][u * 64 + f];
        Te[1][t] = s;
    }
    __syncthreads();
    for (int k = 2; k < 4; ++k) {                    // Tk = 2 L T(k-1) - T(k-2)
        if (t < 640) {
            int v = t >> 6, f = t & 63;
            float s = 0.f;
#pragma unroll
            for (int u = 0; u < 10; ++u) s += Lm[v * 10 + u] * Te[k - 1][u * 64 + f];
            Te[k][t] = 2.f * s - Te[k - 2][t];
        }
        __syncthreads();
    }
    if (t < 320) {                                   // cl4 GEMM + relu
        int v = t >> 5, fo = t & 31;
        float pre = cl4_b[fo];
        for (int f = 0; f < 64; ++f)
#pragma unroll
            for (int k = 0; k < 4; ++k)
                pre += Te[k][v * 64 + f] * cl4_W[fo * 256 + f * 4 + k];
        xg[t] = fmaxf(pre, 0.f);
    }
    __syncthreads();
    if (t < 96) {                                    // maxpool p=3 over v
        int vp = t >> 5, fo = t & 31;
        float m = xg[(vp * 3) * 32 + fo];
        m = fmaxf(m, xg[(vp * 3 + 1) * 32 + fo]);
        m = fmaxf(m, xg[(vp * 3 + 2) * 32 + fo]);
        g4[t] = m;
    }
    __syncthreads();
    if (t < 128) {                                   // fcg1
        float s = fcg1_b[t];
        for (int j = 0; j < 96; ++j) s += fcg1_W[t * 96 + j] * g4[j];
        h1[t] = fmaxf(s, 0.f);
    }
    __syncthreads();
    if (t < 64) {                                    // fcg2 -> g4v
        float s = fcg2_b[t];
        for (int j = 0; j < 128; ++j) s += fcg2_W[t * 128 + j] * h1[j];
        g4v[t] = fmaxf(s, 0.f);
    }
    if (t < 256) {                                   // nn14 (uses emb only)
        float s = nn14_b[t];
        for (int j = 0; j < 640; ++j) s += nn14_W[t * 640 + j] * emb[j];
        hn1[t] = fmaxf(s, 0.f);
    }
    __syncthreads();
    if (t < 64) {                                    // nn24 -> n4v
        float s = nn24_b[t];
        for (int j = 0; j < 256; ++j) s += nn24_W[t * 256 + j] * hn1[j];
        n4v[t] = fmaxf(s, 0.f);
    }
}

// ---------------------------------------------------------------------------
// 6) concat [hidden(512) | nn2(256) | g4(64) | n4(64)] -> data_emb [128,896]
// ---------------------------------------------------------------------------
__global__ void k_concat(const float* __restrict__ hidden, const float* __restrict__ nn2o,
                         const float* __restrict__ g4v, const float* __restrict__ n4v,
                         float* __restrict__ demb) {
    int g = blockIdx.x * 256 + threadIdx.x;
    if (g >= BATCH * SUMFIN) return;
    int b = g / SUMFIN, j = g % SUMFIN;
    float v;
    if (j < 512)      v = hidden[b * 512 + j];
    else if (j < 768) v = nn2o[b * 256 + (j - 512)];
    else if (j < 832) v = g4v[j - 768];
    else              v = n4v[j - 832];
    demb[g] = v;
}

// ---------------------------------------------------------------------------
// 7) row-wise log_softmax over 10 logits
// ---------------------------------------------------------------------------
__global__ void k_logsoftmax(const float* __restrict__ logits, float* __restrict__ logp) {
    int b = threadIdx.x;                             // one block of 128
    float x[OUTC], mx = -3.4e38f;
#pragma unroll
    for (int i = 0; i < OUTC; ++i) { x[i] = logits[b * OUTC + i]; mx = fmaxf(mx, x[i]); }
    float s = 0.f;
#pragma unroll
    for (int i = 0; i < OUTC; ++i) s += __expf(x[i] - mx);
    float ls = __logf(s);
#pragma unroll
    for (int i = 0; i < OUTC; ++i) logp[b * OUTC + i] = x[i] - mx - ls;
}

// ---------------------------------------------------------------------------
extern "C" void kernel_launch(void* const* d_in, const int* in_sizes, int n_in,
                              void* d_out, int out_size, void* d_ws, size_t ws_size,
                              hipStream_t stream) {
    const float* x_in    = (const float*)d_in[0];
    const float* x_embed = (const float*)d_in[1];
    const int*   L_rows  = (const int*)d_in[3];
    const int*   L_cols  = (const int*)d_in[4];
    const float* L_vals  = (const float*)d_in[5];
    const float* L_mg    = (const float*)d_in[6];
    const float* cl1_W = (const float*)d_in[7],  *cl1_b = (const float*)d_in[8];
    const float* fc1_W = (const float*)d_in[9],  *fc1_b = (const float*)d_in[10];
    const float* fc2_W = (const float*)d_in[11], *fc2_b = (const float*)d_in[12];
    const float* fc3_W = (const float*)d_in[13], *fc3_b = (const float*)d_in[14];
    const float* nn1_W = (const float*)d_in[15], *nn1_b = (const float*)d_in[16];
    const float* nn2_W = (const float*)d_in[17], *nn2_b = (const float*)d_in[18];
    const float* cl4_W = (const float*)d_in[19], *cl4_b = (const float*)d_in[20];
    const float* fcg1_W= (const float*)d_in[21], *fcg1_b= (const float*)d_in[22];
    const float* fcg2_W= (const float*)d_in[23], *fcg2_b= (const float*)d_in[24];
    const float* nn14_W= (const float*)d_in[25], *nn14_b= (const float*)d_in[26];
    const float* nn24_W= (const float*)d_in[27], *nn24_b= (const float*)d_in[28];
    const float* fcsum_W=(const float*)d_in[29], *fcsum_b=(const float*)d_in[30];

    // workspace layout (floats)
    float* ws = (float*)d_ws;
    const size_t TSZ = (size_t)D_G * BATCH;          // 2,097,152
    float* T[CL1K];
    for (int k = 0; k < CL1K; ++k) T[k] = ws + k * TSZ;
    float* xp     = ws + CL1K * TSZ;                 // [128, 65536]
    float* fc2o   = xp + (size_t)BATCH * FC1FIN;     // [128, 512]
    float* nn1o   = fc2o + BATCH * 512;              // [128, 512]
    float* nn2o   = nn1o + BATCH * 512;              // [128, 256]
    float* g4v    = nn2o + BATCH * 256;              // [64]
    float* n4v    = g4v + 64;                        // [64]
    float* demb   = n4v + 64;                        // [128, 896]
    float* logits = demb + (size_t)BATCH * SUMFIN;   // [128, 10]

    // d_out layout: decode [128,16384] | hidden [128,512] | logp [128,10]
    float* out_decode = (float*)d_out;
    float* out_hidden = out_decode + (size_t)BATCH * D_G;
    float* out_logp   = out_hidden + (size_t)BATCH * 512;

    // --- Chebyshev chain on gene graph ---
    k_transpose_x0<<<(D_G * BATCH + 255) / 256, 256, 0, stream>>>(x_in, T[0]);
    hipMemsetAsync(T[1], 0, TSZ * sizeof(float), stream);
    k_spmm<<<NNZ_T / SPMM_CHUNK, 128, 0, stream>>>(L_rows, L_cols, L_vals, T[0], T[1]);
    for (int k = 2; k < CL1K; ++k) {
        hipMemsetAsync(T[k], 0, TSZ * sizeof(float), stream);
        k_spmm<<<NNZ_T / SPMM_CHUNK, 128, 0, stream>>>(L_rows, L_cols, L_vals, T[k - 1], T[k]);
        k_cheby_combine<<<(int)((TSZ + 255) / 256), 256, 0, stream>>>(T[k], T[k - 2], (int)TSZ);
    }

    // --- fused cl1 + relu + maxpool ---
    k_cl1_pool<<<(BATCH * VP1 * CL1F) / 256, 256, 0, stream>>>(
        T[0], T[1], T[2], T[3], T[4], T[5], cl1_W, cl1_b, xp);

    // --- dense WMMA GEMMs ---
    // fc1: K=65536, N=512 -> split-K 16 ways (8x16 blocks), atomic reduce
    hipMemsetAsync(out_hidden, 0, (size_t)BATCH * 512 * sizeof(float), stream);
    k_gemm128<<<dim3(512 / 64, 16), 256, 0, stream>>>(xp, fc1_W, fc1_b, out_hidden,
                                                      FC1FIN, 512, 1, 4096, 1);
    k_bias_act<<<(BATCH * 512 + 255) / 256, 256, 0, stream>>>(out_hidden, fc1_b,
                                                              BATCH * 512, 512, 1);
    // fc2: K=512, N=512
    k_gemm128<<<dim3(512 / 64, 1), 256, 0, stream>>>(out_hidden, fc2_W, fc2_b, fc2o,
                                                     512, 512, 1, 512, 0);
    // fc3: K=512, N=16384 (wide -> 256 blocks, no split)
    k_gemm128<<<dim3(D_G / 64, 1), 256, 0, stream>>>(fc2o, fc3_W, fc3_b, out_decode,
                                                     512, D_G, 0, 512, 0);
    // nn1: K=16384, N=512 -> split-K 8 ways
    hipMemsetAsync(nn1o, 0, (size_t)BATCH * 512 * sizeof(float), stream);
    k_gemm128<<<dim3(512 / 64, 8), 256, 0, stream>>>(x_in, nn1_W, nn1_b, nn1o,
                                                     D_G, 512, 1, 2048, 1);
    k_bias_act<<<(BATCH * 512 + 255) / 256, 256, 0, stream>>>(nn1o, nn1_b,
                                                              BATCH * 512, 512, 1);
    // nn2: K=512, N=256
    k_gemm128<<<dim3(256 / 64, 1), 256, 0, stream>>>(nn1o, nn2_W, nn2_b, nn2o,
                                                     512, 256, 1, 512, 0);

    // --- GCN4 + NN4 (batch-invariant) ---
    k_gcn4<<<1, 256, 0, stream>>>(x_embed, L_mg, cl4_W, cl4_b, fcg1_W, fcg1_b,
                                  fcg2_W, fcg2_b, nn14_W, nn14_b, nn24_W, nn24_b,
                                  g4v, n4v);

    // --- concat + summary head + log_softmax ---
    k_concat<<<(BATCH * SUMFIN + 255) / 256, 256, 0, stream>>>(out_hidden, nn2o, g4v, n4v, demb);
    k_gemm128<<<dim3(1, 1), 256, 0, stream>>>(demb, fcsum_W, fcsum_b, logits,
                                              SUMFIN, OUTC, 0, SUMFIN, 0);
    k_logsoftmax<<<1, 128, 0, stream>>>(logits, out_logp);
}